// SPUTransformer_37254546326064
// MI455X (gfx1250) — compile-verified
//
#include <hip/hip_runtime.h>

// Streaming elementwise SPU forward transform.
// N=8192 rows, 40 B traffic/row -> HBM-roofline time ~14 ns: launch-bound.
// No matrix structure -> WMMA/TDM intentionally not used (would only add
// data movement). Optimal CDNA5 shape: 1 thread/row, wave32 blocks of 256,
// coalesced b64 vector loads/stores with non-temporal hints, hardware
// v_exp_f32 / v_rcp_f32 for the sigmoid branch, branchless selects.

typedef float v2f __attribute__((ext_vector_type(2)));

__device__ __forceinline__ float spu_eval(float x) {
    // x >= 0 : x*x - 0.5
    // x <  0 : sigmoid(-x) - 1 == 1/(1 + e^x) - 1
    float p = __fmaf_rn(x, x, -0.5f);
    float e = __expf(x);                               // v_exp_f32 path
    float s = __builtin_amdgcn_rcpf(1.0f + e) - 1.0f;  // v_rcp_f32
    return (x >= 0.0f) ? p : s;
}

__global__ __launch_bounds__(256)
void spu_transformer_fwd(const v2f* __restrict__ bounds,
                         const v2f* __restrict__ last_slopes,
                         const v2f* __restrict__ last_shifts,
                         const v2f* __restrict__ last_bounds,
                         v2f* __restrict__ out, int n)
{
    int i = blockIdx.x * blockDim.x + threadIdx.x;
    if (i >= n) return;

    // Non-temporal b64 loads: pure streaming data, read exactly once.
    v2f b   = __builtin_nontemporal_load(bounds      + i);
    v2f ls  = __builtin_nontemporal_load(last_slopes + i);
    v2f lsh = __builtin_nontemporal_load(last_shifts + i);
    v2f lbd = __builtin_nontemporal_load(last_bounds + i);

    float l = b.x, u = b.y;
    float vl = spu_eval(l);
    float vu = spu_eval(u);

    float diff = u - l;
    bool  neg   = (u <= 0.0f);
    bool  pos   = (l >= 0.0f);
    bool  cross = !(neg || pos);

    // Keep the IEEE divide so u==l propagates inf/nan exactly like the ref.
    float all_slopes = (vu - vl) / diff;
    float slope_u = (pos || cross) ? all_slopes : 0.0f;
    float slope_l = neg            ? all_slopes : 0.0f;

    bool  sw  = (all_slopes < 0.0f);
    float v_l = sw ? vu : vl;
    float v_u = sw ? vl : vu;
    float b_l = sw ? u  : l;
    float b_u = sw ? l  : u;

    float bnd_l = cross ? -0.5f : v_l;
    float bnd_u = v_u;

    float shift_u = __fmaf_rn(-slope_u, b_u, v_u);
    float shift_l = cross ? -0.5f : __fmaf_rn(-slope_l, b_l, v_l);

    float Ud = slope_u * ls.y;
    float UV = __fmaf_rn(slope_u, lsh.y, shift_u);
    float Ld = slope_l * ls.x;
    float LV = __fmaf_rn(slope_l, lsh.x, shift_l);

    // lb = max(Ld,0)*lbd.x + min(Ld,0)*lbd.y + LV  (FMA chain)
    float lb = __fmaf_rn(fmaxf(Ld, 0.0f), lbd.x,
               __fmaf_rn(fminf(Ld, 0.0f), lbd.y, LV));
    // ub = max(Ud,0)*lbd.y + min(Ud,0)*lbd.x + UV
    float ub = __fmaf_rn(fmaxf(Ud, 0.0f), lbd.y,
               __fmaf_rn(fminf(Ud, 0.0f), lbd.x, UV));

    float out_l = (lb > bnd_l) ? lb : bnd_l;
    float out_u = (ub < bnd_u) ? ub : bnd_u;

    v2f r;
    r.x = out_l;
    r.y = out_u;
    __builtin_nontemporal_store(r, out + i);
}

extern "C" void kernel_launch(void* const* d_in, const int* in_sizes, int n_in,
                              void* d_out, int out_size, void* d_ws, size_t ws_size,
                              hipStream_t stream) {
    const v2f* bounds      = (const v2f*)d_in[0];
    const v2f* last_slopes = (const v2f*)d_in[1];
    const v2f* last_shifts = (const v2f*)d_in[2];
    const v2f* last_bounds = (const v2f*)d_in[3];
    v2f*       out         = (v2f*)d_out;

    const int n = in_sizes[0] / 2;           // rows (bounds is N x 2)
    const int block = 256;                   // 8 wave32 waves / workgroup
    const int grid  = (n + block - 1) / block;

    spu_transformer_fwd<<<grid, block, 0, stream>>>(
        bounds, last_slopes, last_shifts, last_bounds, out, n);
}